// OptimizedODEFunc_10033043604046
// MI455X (gfx1250) — compile-verified
//
#include <hip/hip_runtime.h>

#define N_NODES 20000
#define N_EDGES 320000
#define D 128
#define B 4
#define LN_EPS 1e-5f
#define MAX_NORM 10.0f
#define TSTR 132   // padded LDS row stride (floats); conflict-free A-frag reads, 16B-aligned rows
#define NTILES (B * (N_NODES / 16))   // 5000
#define DYN_GRID 625                  // 8 tiles per persistent block

typedef __attribute__((ext_vector_type(2))) float v2f;
typedef __attribute__((ext_vector_type(8))) float v8f;

__device__ __forceinline__ float ntn(float v) {
  if (v != v) return 0.0f;                                   // NaN -> 0
  return fminf(fmaxf(v, -3.402823466e38f), 3.402823466e38f); // +/-inf -> +/-FLT_MAX
}

__device__ __forceinline__ void atomicMaxF(float* addr, float val) {
  int old = __float_as_int(*addr);
  while (__int_as_float(old) < val) {
    int prev = atomicCAS((int*)addr, old, __float_as_int(val));
    if (prev == old) break;
    old = prev;
  }
}

// ---- CDNA5 async global->LDS (ASYNCcnt-tracked), per-lane 16B ----
__device__ __forceinline__ void async_b128(const float* g, float* l) {
  unsigned int lds = (unsigned int)(uintptr_t)l;          // low 32 bits = LDS offset
  unsigned long long ga = (unsigned long long)(uintptr_t)g;
  asm volatile("global_load_async_to_lds_b128 %0, %1, off" :: "v"(lds), "v"(ga) : "memory");
}
__device__ __forceinline__ void wait_async0() {
  asm volatile("s_wait_asynccnt 0" ::: "memory");
}

__global__ void init_maxv_kernel(float* maxv) {
  if (threadIdx.x < B) maxv[threadIdx.x] = -3.402823466e38f;
}

__global__ void deg_kernel(const int* __restrict__ src, float* __restrict__ deg) {
  int e = blockIdx.x * blockDim.x + threadIdx.x;
  if (e < N_EDGES)
    __hip_atomic_fetch_add(&deg[src[e]], 1.0f, __ATOMIC_RELAXED, __HIP_MEMORY_SCOPE_AGENT);
}

// logits[b,n] = dot(nan_to_num(x[b,n,:]), Wa) + ba ; track per-batch max
__global__ void logits_kernel(const float* __restrict__ x, const float* __restrict__ Wa,
                              const float* __restrict__ ba,
                              float* __restrict__ logits, float* __restrict__ maxv) {
  int row = blockIdx.x * 8 + (threadIdx.x >> 5);  // flat (b*N + n)
  if (row >= B * N_NODES) return;
  int lane = threadIdx.x & 31;
  const float4* xr = (const float4*)(x + (size_t)row * D);
  const float4* wr = (const float4*)Wa;
  float4 xv = xr[lane];
  float4 wv = wr[lane];
  float s = ntn(xv.x) * wv.x + ntn(xv.y) * wv.y + ntn(xv.z) * wv.z + ntn(xv.w) * wv.w;
  #pragma unroll
  for (int o = 16; o > 0; o >>= 1) s += __shfl_xor(s, o, 32);
  if (lane == 0) {
    float l = s + ba[0];
    logits[row] = l;
    atomicMaxF(&maxv[row / N_NODES], l);
  }
}

__global__ void expsum_kernel(float* __restrict__ logits, const float* __restrict__ maxv,
                              float* __restrict__ sumv) {
  int i = blockIdx.x * blockDim.x + threadIdx.x;
  if (i >= B * N_NODES) return;
  float e = expf(logits[i] - maxv[i / N_NODES]);
  logits[i] = e;
  __hip_atomic_fetch_add(&sumv[i / N_NODES], e, __ATOMIC_RELAXED, __HIP_MEMORY_SCOPE_AGENT);
}

__global__ void attn_kernel(const float* __restrict__ logits, const float* __restrict__ sumv,
                            float* __restrict__ attn) {
  int i = blockIdx.x * blockDim.x + threadIdx.x;
  if (i >= B * N_NODES) return;
  attn[i] = logits[i] / sumv[i / N_NODES];
}

// diffusion[b, src[e], :] += (attn[b,dst[e]] / (deg[src[e]]+1e-8)) * nan_to_num(x[b, dst[e], :])
// one wave per (edge, batch); lane covers 4 channels, L2-resident f32 atomics.
__global__ void spmm_kernel(const float* __restrict__ x, const int* __restrict__ src,
                            const int* __restrict__ dst, const float* __restrict__ deg,
                            const float* __restrict__ attn, float* __restrict__ diff) {
  int gw = blockIdx.x * 8 + (threadIdx.x >> 5);
  if (gw >= N_EDGES * B) return;
  int lane = threadIdx.x & 31;
  int e = gw >> 2, b = gw & 3;
  int s = src[e], d = dst[e];
  float w = attn[(size_t)b * N_NODES + d] / (deg[s] + 1e-8f);
  const float4* xr = (const float4*)(x + ((size_t)b * N_NODES + d) * D);
  float4 v = xr[lane];
  float* o = diff + ((size_t)b * N_NODES + s) * D + lane * 4;
  __hip_atomic_fetch_add(o + 0, ntn(v.x) * w, __ATOMIC_RELAXED, __HIP_MEMORY_SCOPE_AGENT);
  __hip_atomic_fetch_add(o + 1, ntn(v.y) * w, __ATOMIC_RELAXED, __HIP_MEMORY_SCOPE_AGENT);
  __hip_atomic_fetch_add(o + 2, ntn(v.z) * w, __ATOMIC_RELAXED, __HIP_MEMORY_SCOPE_AGENT);
  __hip_atomic_fetch_add(o + 3, ntn(v.w) * w, __ATOMIC_RELAXED, __HIP_MEMORY_SCOPE_AGENT);
}

// Persistent fused kernel: W1 & W2 stay LDS-resident; each block loops over 8 node-tiles.
// Per tile: async x->LDS (prefetched) -> GEMM1(WMMA f32 16x16x4) -> LN -> SiLU
//           -> GEMM2 -> tanh -> +diffusion -> norm clip -> store.
__global__ __launch_bounds__(256)
void dyn_kernel(const float* __restrict__ x, const float* __restrict__ W1,
                const float* __restrict__ b1, const float* __restrict__ gamma,
                const float* __restrict__ beta, const float* __restrict__ W2,
                const float* __restrict__ b2, const float* __restrict__ diff,
                const float* __restrict__ dscale, const float* __restrict__ tscale,
                float* __restrict__ out) {
  __shared__ float sW1[D * D];     // 64KB, resident
  __shared__ float sW2[D * D];     // 64KB, resident
  __shared__ float sA[16 * TSTR];  // x tile (async target)
  __shared__ float sH[16 * TSTR];  // h / dynamics tile

  int t = threadIdx.x;
  int wave = t >> 5, lane = t & 31;
  int half = lane >> 4, l16 = lane & 15;
  int c0 = wave * 16;
  int row = t >> 4, li = t & 15;

  // ---- async-stage both weight matrices once (32 x B128 per thread-pair of matrices) ----
  #pragma unroll
  for (int i = 0; i < 16; i++) async_b128(W1 + (i * 256 + t) * 4, sW1 + (i * 256 + t) * 4);
  #pragma unroll
  for (int i = 0; i < 16; i++) async_b128(W2 + (i * 256 + t) * 4, sW2 + (i * 256 + t) * 4);

  // hoisted per-thread constants
  float bias1 = b1[c0 + l16];
  float bias2 = b2[c0 + l16];
  float ds = dscale[0], ts = tscale[0];
  float gam[8], bet[8];
  #pragma unroll
  for (int j = 0; j < 8; j++) { gam[j] = gamma[li + 16 * j]; bet[j] = beta[li + 16 * j]; }

  // ---- prefetch first x tile ----
  int tile = blockIdx.x;
  {
    int b = tile / (N_NODES / 16), n0 = (tile % (N_NODES / 16)) * 16;
    const float* xb = x + ((size_t)b * N_NODES + n0) * D;
    #pragma unroll
    for (int i = 0; i < 2; i++) {
      int fi = i * 256 + t, r = fi >> 5, c4 = fi & 31;
      async_b128(xb + fi * 4, sA + r * TSTR + c4 * 4);
    }
  }

  for (; tile < NTILES; tile += DYN_GRID) {
    int b = tile / (N_NODES / 16), node0 = (tile % (N_NODES / 16)) * 16;

    wait_async0();      // weights (first iter) + current x tile landed
    __syncthreads();

    // nan_to_num fixup in LDS
    #pragma unroll
    for (int j = 0; j < 8; j++) {
      int c = li + 16 * j;
      sA[row * TSTR + c] = ntn(sA[row * TSTR + c]);
    }
    __syncthreads();

    // ---- GEMM 1: h = x @ W1 ----
    v8f acc = {};
    #pragma unroll
    for (int k0 = 0; k0 < D; k0 += 4) {
      int k = k0 + 2 * half;
      v2f a, bb;
      a[0] = sA[l16 * TSTR + k];
      a[1] = sA[l16 * TSTR + k + 1];
      bb[0] = sW1[k * D + c0 + l16];
      bb[1] = sW1[(k + 1) * D + c0 + l16];
      acc = __builtin_amdgcn_wmma_f32_16x16x4_f32(false, a, false, bb, (short)0, acc, false, false);
    }
    __syncthreads();  // all waves done reading sA

    // prefetch next tile's x into sA (overlaps LN/GEMM2/combine)
    int nt = tile + DYN_GRID;
    if (nt < NTILES) {
      int nb2 = nt / (N_NODES / 16), nn0 = (nt % (N_NODES / 16)) * 16;
      const float* xb = x + ((size_t)nb2 * N_NODES + nn0) * D;
      #pragma unroll
      for (int i = 0; i < 2; i++) {
        int fi = i * 256 + t, r = fi >> 5, c4 = fi & 31;
        async_b128(xb + fi * 4, sA + r * TSTR + c4 * 4);
      }
    }

    // store h tile
    #pragma unroll
    for (int i = 0; i < 8; i++) sH[(i + 8 * half) * TSTR + c0 + l16] = acc[i] + bias1;
    __syncthreads();

    // ---- LayerNorm + SiLU, 16-thread row teams (in-wave shfl reduction) ----
    {
      float sum = 0.f, sq = 0.f, vals[8];
      #pragma unroll
      for (int j = 0; j < 8; j++) {
        float v = sH[row * TSTR + li + 16 * j];
        vals[j] = v; sum += v; sq += v * v;
      }
      #pragma unroll
      for (int o = 8; o > 0; o >>= 1) { sum += __shfl_xor(sum, o, 16); sq += __shfl_xor(sq, o, 16); }
      float mu = sum * (1.0f / D);
      float var = sq * (1.0f / D) - mu * mu;
      float rstd = rsqrtf(var + LN_EPS);
      #pragma unroll
      for (int j = 0; j < 8; j++) {
        float v = (vals[j] - mu) * rstd * gam[j] + bet[j];
        v = v / (1.0f + expf(-v));  // SiLU
        sH[row * TSTR + li + 16 * j] = v;
      }
    }
    __syncthreads();

    // ---- GEMM 2: dynamics = tanh(h @ W2 + b2) ----
    v8f acc2 = {};
    #pragma unroll
    for (int k0 = 0; k0 < D; k0 += 4) {
      int k = k0 + 2 * half;
      v2f a, bb;
      a[0] = sH[l16 * TSTR + k];
      a[1] = sH[l16 * TSTR + k + 1];
      bb[0] = sW2[k * D + c0 + l16];
      bb[1] = sW2[(k + 1) * D + c0 + l16];
      acc2 = __builtin_amdgcn_wmma_f32_16x16x4_f32(false, a, false, bb, (short)0, acc2, false, false);
    }
    __syncthreads();
    #pragma unroll
    for (int i = 0; i < 8; i++)
      sH[(i + 8 * half) * TSTR + c0 + l16] = tanhf(acc2[i] + bias2);
    __syncthreads();

    // ---- combine with diffusion, norm-clip, write ----
    {
      const float* dfb = diff + ((size_t)b * N_NODES + node0) * D;
      float dx[8], sq = 0.f;
      #pragma unroll
      for (int j = 0; j < 8; j++) {
        int c = li + 16 * j;
        float v = ts * (sH[row * TSTR + c] + dfb[row * D + c] * ds);
        dx[j] = v; sq += v * v;
      }
      #pragma unroll
      for (int o = 8; o > 0; o >>= 1) sq += __shfl_xor(sq, o, 16);
      float norm = sqrtf(sq);
      float scale = fminf(MAX_NORM / (norm + 1e-8f), 1.0f);
      float* ob = out + ((size_t)b * N_NODES + node0) * D;
      #pragma unroll
      for (int j = 0; j < 8; j++) {
        int c = li + 16 * j;
        ob[row * D + c] = ntn(dx[j] * scale);
      }
    }
  }
}

extern "C" void kernel_launch(void* const* d_in, const int* in_sizes, int n_in,
                              void* d_out, int out_size, void* d_ws, size_t ws_size,
                              hipStream_t stream) {
  const float* x     = (const float*)d_in[1];
  const int*   esrc  = (const int*)d_in[2];
  const int*   edst  = (const int*)d_in[3];
  const float* W1    = (const float*)d_in[4];
  const float* b1    = (const float*)d_in[5];
  const float* gamma = (const float*)d_in[6];
  const float* beta  = (const float*)d_in[7];
  const float* W2    = (const float*)d_in[8];
  const float* b2    = (const float*)d_in[9];
  const float* Wa    = (const float*)d_in[10];
  const float* ba    = (const float*)d_in[11];
  const float* dsc   = (const float*)d_in[12];
  const float* tsc   = (const float*)d_in[13];
  float* out = (float*)d_out;

  float* ws     = (float*)d_ws;
  float* diff   = ws;                                 // B*N*D
  float* deg    = diff + (size_t)B * N_NODES * D;     // N
  float* logits = deg + N_NODES;                      // B*N
  float* attn   = logits + (size_t)B * N_NODES;       // B*N
  float* maxv   = attn + (size_t)B * N_NODES;         // B
  float* sumv   = maxv + B;                           // B
  size_t totalf = (size_t)B * N_NODES * D + N_NODES + 2 * (size_t)B * N_NODES + 2 * B;

  hipMemsetAsync(ws, 0, totalf * sizeof(float), stream);
  init_maxv_kernel<<<1, 32, 0, stream>>>(maxv);
  deg_kernel<<<(N_EDGES + 255) / 256, 256, 0, stream>>>(esrc, deg);
  logits_kernel<<<(B * N_NODES + 7) / 8, 256, 0, stream>>>(x, Wa, ba, logits, maxv);
  expsum_kernel<<<(B * N_NODES + 255) / 256, 256, 0, stream>>>(logits, maxv, sumv);
  attn_kernel<<<(B * N_NODES + 255) / 256, 256, 0, stream>>>(logits, sumv, attn);
  spmm_kernel<<<(N_EDGES * B + 7) / 8, 256, 0, stream>>>(x, esrc, edst, deg, attn, diff);
  dyn_kernel<<<DYN_GRID, 256, 0, stream>>>(x, W1, b1, gamma, beta, W2, b2,
                                           diff, dsc, tsc, out);
}